// SymmetryLoss_80831284510870
// MI455X (gfx1250) — compile-verified
//
#include <hip/hip_runtime.h>

// SymmetryLoss: out[b,kp] = (x[b,kp] - x[b,sym[kp]])^2 over B x 15 float32.
// sym: kp 0..2 -> self (output 0); pairs (3,4)(5,6)(7,8)(9,10)(11,12)(13,14)
// share one squared difference.
//
// Memory-bound streaming problem: 120 MB in + 120 MB out => ~10 us at
// 23.3 TB/s HBM. Strategy: 4 rows = 60 floats = 240 B = 15 aligned 16-byte
// chunks per thread, moved with b128 non-temporal loads/stores. All offsets
// fit in 32 bits (max 120 MB), enabling saddr + 32-bit voffset addressing.

typedef float v4f __attribute__((ext_vector_type(4)));

__global__ __launch_bounds__(256)
void symmetry_loss_vec4_kernel(const float* __restrict__ x,
                               float* __restrict__ out,
                               int ngroups,   // full groups of 4 rows
                               int nrows)     // total rows
{
    const int g = blockIdx.x * blockDim.x + threadIdx.x;

    if (g < ngroups) {
        const unsigned off = (unsigned)g * 60u;          // 32-bit element offset
        const v4f* __restrict__ src = reinterpret_cast<const v4f*>(x   + off);
        v4f*       __restrict__ dst = reinterpret_cast<v4f*>(out + off);

        // 240 B as 15 x global_load_b128 (non-temporal, touch-once stream).
        float v[60];
#pragma unroll
        for (int i = 0; i < 15; ++i) {
            v4f t = __builtin_nontemporal_load(&src[i]);
            v[4 * i + 0] = t.x;
            v[4 * i + 1] = t.y;
            v[4 * i + 2] = t.z;
            v[4 * i + 3] = t.w;
        }

        // In-place compute over 4 rows of 15.
#pragma unroll
        for (int r = 0; r < 4; ++r) {
            const int base = 15 * r;
            v[base + 0] = 0.0f;
            v[base + 1] = 0.0f;
            v[base + 2] = 0.0f;
#pragma unroll
            for (int p = 3; p < 15; p += 2) {
                float d = v[base + p] - v[base + p + 1];
                d = d * d;
                v[base + p]     = d;
                v[base + p + 1] = d;
            }
        }

        // 240 B as 15 x global_store_b128 (non-temporal).
#pragma unroll
        for (int i = 0; i < 15; ++i) {
            v4f o;
            o.x = v[4 * i + 0];
            o.y = v[4 * i + 1];
            o.z = v[4 * i + 2];
            o.w = v[4 * i + 3];
            __builtin_nontemporal_store(o, &dst[i]);
        }
        return;
    }

    // Tail rows (nrows % 4 != 0): thread g in [ngroups, ngroups+tail) handles
    // one leftover row scalar-wise. Dead code for B divisible by 4.
    const int tail_row = 4 * ngroups + (g - ngroups);
    if (tail_row < nrows) {
        const unsigned roff = (unsigned)tail_row * 15u;
        const float* __restrict__ row = x   + roff;
        float*       __restrict__ o   = out + roff;
        o[0] = 0.0f; o[1] = 0.0f; o[2] = 0.0f;
#pragma unroll
        for (int p = 3; p < 15; p += 2) {
            float d = row[p] - row[p + 1];
            d = d * d;
            o[p]     = d;
            o[p + 1] = d;
        }
    }
}

extern "C" void kernel_launch(void* const* d_in, const int* in_sizes, int n_in,
                              void* d_out, int out_size, void* d_ws, size_t ws_size,
                              hipStream_t stream)
{
    const float* x   = (const float*)d_in[0];
    float*       out = (float*)d_out;

    const int total   = in_sizes[0];        // B * 15 (30M, fits int)
    const int nrows   = total / 15;
    const int ngroups = nrows / 4;
    const int tail    = nrows - 4 * ngroups;

    const int threads = 256;                // 8 wave32 waves per block
    int work = ngroups + tail;              // group threads + tail-row threads
    if (work < 1) work = 1;
    const int blocks = (work + threads - 1) / threads;

    symmetry_loss_vec4_kernel<<<blocks, threads, 0, stream>>>(x, out, ngroups, nrows);
}